// UnifiedLossFunction_77996606095838
// MI455X (gfx1250) — compile-verified
//
#include <hip/hip_runtime.h>
#include <math.h>

#define NB    40
#define NPAD  48
#define ZMINF 1e-6f
#define ZMAXF 2.0f
#define LAMF  0.5f
#define EPSF  1e-8f
#define MAXIT 50

typedef float v2f __attribute__((ext_vector_type(2)));
typedef float v4f __attribute__((ext_vector_type(4)));
typedef float v8f __attribute__((ext_vector_type(8)));

static __device__ __forceinline__ v8f wmma4(v2f a, v2f b, v8f c) {
  // V_WMMA_F32_16X16X4_F32 : D(16x16,f32) = A(16x4,f32) * B(4x16,f32) + C
  return __builtin_amdgcn_wmma_f32_16x16x4_f32(false, a, false, b, (short)0, c, false, false);
}

// ---------------- kernel 1: K and K*C matrices (48x48, zero padded) ----------------
__global__ void k_setup(float* __restrict__ Kg, float* __restrict__ KCg) {
  int idx = blockIdx.x * blockDim.x + threadIdx.x;
  if (idx >= NPAD * NPAD) return;
  int i = idx / NPAD, j = idx % NPAD;
  float kv = 0.f, kcv = 0.f;
  if (i < NB && j < NB) {
    const float step = (ZMAXF - ZMINF) / (float)(NB - 1); // cost centers: linspace(zmin,zmax,40)
    float d = (float)(i - j) * step;
    float c = d * d + 1.0f;                               // (ci-cj)^2 + DELTA^2
    float e = __expf(fminf(fmaxf(-c / LAMF, -100.f), 100.f));
    kv = e; kcv = e * c;
  }
  Kg[idx] = kv; KCg[idx] = kcv;
}

// ---------------- kernel 2: per-block stats partials over g_r ----------------
__global__ void k_stats1(const float* __restrict__ cf, int B, float* __restrict__ bpart) {
  __shared__ float sc[256], ss[256], sq[256];
  int tid = threadIdx.x;
  int idx = blockIdx.x * 256 + tid;
  float c = 0.f, s = 0.f, q = 0.f;
  if (idx < B) {
    float g = cf[idx * 4 + 1];
    if (g >= 0.2f && g <= 1.5f) { c = 1.f; s = g; q = g * g; }
  }
  sc[tid] = c; ss[tid] = s; sq[tid] = q;
  __syncthreads();
  for (int o = 128; o > 0; o >>= 1) {
    if (tid < o) { sc[tid] += sc[tid + o]; ss[tid] += ss[tid + o]; sq[tid] += sq[tid + o]; }
    __syncthreads();
  }
  if (tid == 0) {
    bpart[blockIdx.x * 3 + 0] = sc[0];
    bpart[blockIdx.x * 3 + 1] = ss[0];
    bpart[blockIdx.x * 3 + 2] = sq[0];
  }
}

// ---------------- kernel 3: finish stats -> mean, offset, has_valid ----------------
__global__ void k_stats2(const float* __restrict__ bpart, int nb1, float* __restrict__ stats) {
  __shared__ float sc[256], ss[256], sq[256];
  int tid = threadIdx.x;
  float c = 0.f, s = 0.f, q = 0.f;
  for (int i = tid; i < nb1; i += 256) {
    c += bpart[3 * i]; s += bpart[3 * i + 1]; q += bpart[3 * i + 2];
  }
  sc[tid] = c; ss[tid] = s; sq[tid] = q;
  __syncthreads();
  for (int o = 128; o > 0; o >>= 1) {
    if (tid < o) { sc[tid] += sc[tid + o]; ss[tid] += ss[tid + o]; sq[tid] += sq[tid + o]; }
    __syncthreads();
  }
  if (tid == 0) {
    float cnt = sc[0], sum = ss[0], sumsq = sq[0];
    float mean = sum / fmaxf(cnt, 1.f);
    float ssd  = fmaxf(sumsq - 2.f * mean * sum + mean * mean * cnt, 0.f);
    float var  = ssd / fmaxf(cnt - 1.f, 1.f);
    float std  = (cnt > 1.f) ? sqrtf(var) : 0.1f;
    float off  = (std > 0.1f) ? 0.3f * std : 0.1f;
    stats[0] = mean; stats[1] = off; stats[2] = (cnt > 0.f) ? 1.f : 0.f;
  }
}

// ---------------- kernel 4: main Sinkhorn via WMMA f32 16x16x4 ----------------
// one wave32 handles 16 samples; 8 waves / block; private LDS slice per wave
__global__ void __launch_bounds__(256)
k_main(const float* __restrict__ pred, const float* __restrict__ target,
       const float* __restrict__ cf, const float* __restrict__ Kg,
       const float* __restrict__ KCg, const float* __restrict__ stats,
       float* __restrict__ wpart) {
  __shared__ __align__(16) float lds[8][2][NPAD * 16];
  const int l    = threadIdx.x & 31;
  const int wib  = threadIdx.x >> 5;
  const int gw   = blockIdx.x * 8 + wib;
  const int sbase = gw * 16;
  const int sl   = l & 15;
  const int h    = l >> 4;
  float* R0 = &lds[wib][0][0];   // staging for a / u-relayout / final row sums
  float* R1 = &lds[wib][1][0];   // staging for b / v-relayout

  // ---- build distributions a (lanes 0-15 from pred) and b (lanes 16-31 from target),
  //      stored column-major [bin*16 + sample] so D-layout loads are contiguous ----
  {
    float x = (h == 0) ? pred[sbase + sl] : target[sbase + sl];
    float* dst = (h == 0) ? R0 : R1;
    const float bw = (ZMAXF - ZMINF) / (float)NB;
    const float sg = bw * 0.5f;
    const float inv2s2 = 1.0f / (2.0f * sg * sg);
    float ssum = 0.f;
#pragma unroll
    for (int n = 0; n < NB; n++) {
      float d = x - (ZMINF + ((float)n + 0.5f) * bw);
      ssum += __expf(-d * d * inv2s2);
    }
    float inv = 1.0f / (ssum + EPSF);
#pragma unroll
    for (int n = 0; n < NB; n++) {
      float d = x - (ZMINF + ((float)n + 0.5f) * bw);
      dst[n * 16 + sl] = __expf(-d * d * inv2s2) * inv;
    }
#pragma unroll
    for (int n = NB; n < NPAD; n++) dst[n * 16 + sl] = 0.f;
  }

  // ---- load a,b in WMMA C/D layout: VGPR r holds rows r+8h, col sl+16t ----
  float aD[3][8], bD[3][8];
#pragma unroll
  for (int t = 0; t < 3; t++) {
    int base = (sl + 16 * t) * 16 + 8 * h;
    v4f lo = *(const v4f*)&R0[base];
    v4f hi = *(const v4f*)&R0[base + 4];
#pragma unroll
    for (int r = 0; r < 4; r++) { aD[t][r] = lo[r]; aD[t][r + 4] = hi[r]; }
    lo = *(const v4f*)&R1[base];
    hi = *(const v4f*)&R1[base + 4];
#pragma unroll
    for (int r = 0; r < 4; r++) { bD[t][r] = lo[r]; bD[t][r + 4] = hi[r]; }
  }

  // ---- K (symmetric, padded) resident in B-operand layout registers ----
  // B chunk (k-rows 4c..4c+3, cols 16t..16t+15): lane l -> col 16t+sl,
  // VGPR0 = row 4c+2h, VGPR1 = row 4c+2h+1
  float kB[12][3][2];
#pragma unroll
  for (int c = 0; c < 12; c++)
#pragma unroll
    for (int t = 0; t < 3; t++) {
      int i0 = (4 * c + 2 * h) * NPAD + 16 * t + sl;
      kB[c][t][0] = Kg[i0];
      kB[c][t][1] = Kg[i0 + NPAD];
    }

  // ---- Sinkhorn state: u,v in A-operand layout (lane=sample sl, VGPR0=bin 4c+2h) ----
  float uA[12][2], vA[12][2];
#pragma unroll
  for (int c = 0; c < 12; c++) { vA[c][0] = 1.f; vA[c][1] = 1.f; }

#pragma unroll 1
  for (int it = 0; it < MAXIT; ++it) {
    // ---- u = a / (v @ K + eps) ----
    v8f acc[3] = { {0,0,0,0,0,0,0,0}, {0,0,0,0,0,0,0,0}, {0,0,0,0,0,0,0,0} };
#pragma unroll
    for (int c = 0; c < 12; c++) {
      v2f av; av[0] = vA[c][0]; av[1] = vA[c][1];
#pragma unroll
      for (int t = 0; t < 3; t++) {
        v2f bv; bv[0] = kB[c][t][0]; bv[1] = kB[c][t][1];
        acc[t] = wmma4(av, bv, acc[t]);
      }
    }
#pragma unroll
    for (int t = 0; t < 3; t++) {
      v4f p0, p1;
#pragma unroll
      for (int r = 0; r < 4; r++) {
        p0[r] = aD[t][r]     * __builtin_amdgcn_rcpf(acc[t][r]     + EPSF);
        p1[r] = aD[t][r + 4] * __builtin_amdgcn_rcpf(acc[t][r + 4] + EPSF);
      }
      int base = (sl + 16 * t) * 16 + 8 * h;
      *(v4f*)&R0[base]     = p0;
      *(v4f*)&R0[base + 4] = p1;
    }
#pragma unroll
    for (int c = 0; c < 12; c++) {           // D-layout -> A-layout via LDS
      int k0 = 4 * c + 2 * h;
      uA[c][0] = R0[k0 * 16 + sl];
      uA[c][1] = R0[(k0 + 1) * 16 + sl];
    }

    // ---- v = b / (u @ K + eps) ----
    v8f acd[3] = { {0,0,0,0,0,0,0,0}, {0,0,0,0,0,0,0,0}, {0,0,0,0,0,0,0,0} };
#pragma unroll
    for (int c = 0; c < 12; c++) {
      v2f av; av[0] = uA[c][0]; av[1] = uA[c][1];
#pragma unroll
      for (int t = 0; t < 3; t++) {
        v2f bv; bv[0] = kB[c][t][0]; bv[1] = kB[c][t][1];
        acd[t] = wmma4(av, bv, acd[t]);
      }
    }
#pragma unroll
    for (int t = 0; t < 3; t++) {
      v4f p0, p1;
#pragma unroll
      for (int r = 0; r < 4; r++) {
        p0[r] = bD[t][r]     * __builtin_amdgcn_rcpf(acd[t][r]     + EPSF);
        p1[r] = bD[t][r + 4] * __builtin_amdgcn_rcpf(acd[t][r + 4] + EPSF);
      }
      int base = (sl + 16 * t) * 16 + 8 * h;
      *(v4f*)&R1[base]     = p0;
      *(v4f*)&R1[base + 4] = p1;
    }
#pragma unroll
    for (int c = 0; c < 12; c++) {
      int k0 = 4 * c + 2 * h;
      vA[c][0] = R1[k0 * 16 + sl];
      vA[c][1] = R1[(k0 + 1) * 16 + sl];
    }
  }

  // ---- cost = u @ (K*C) @ v ; KC B-operands streamed from global (single use) ----
  v8f e[3] = { {0,0,0,0,0,0,0,0}, {0,0,0,0,0,0,0,0}, {0,0,0,0,0,0,0,0} };
#pragma unroll
  for (int c = 0; c < 12; c++) {
    v2f av; av[0] = uA[c][0]; av[1] = uA[c][1];
#pragma unroll
    for (int t = 0; t < 3; t++) {
      int i0 = (4 * c + 2 * h) * NPAD + 16 * t + sl;
      v2f bv; bv[0] = KCg[i0]; bv[1] = KCg[i0 + NPAD];
      e[t] = wmma4(av, bv, e[t]);
    }
  }
  // reload final v in D-layout from R1 and form per-lane partial dot
  float p[8];
#pragma unroll
  for (int r = 0; r < 8; r++) p[r] = 0.f;
#pragma unroll
  for (int t = 0; t < 3; t++) {
    int base = (sl + 16 * t) * 16 + 8 * h;
    v4f lo = *(const v4f*)&R1[base];
    v4f hi = *(const v4f*)&R1[base + 4];
#pragma unroll
    for (int r = 0; r < 4; r++) {
      p[r]     += e[t][r]     * lo[r];
      p[r + 4] += e[t][r + 4] * hi[r];
    }
  }
  // transpose partials through LDS: [sample*16 + col-lane]
#pragma unroll
  for (int r = 0; r < 8; r++) R0[(r + 8 * h) * 16 + sl] = p[r];

  float whk = 0.f, ww = 0.f;
  if (l < 16) {
    float cost = 0.f;
#pragma unroll
    for (int j = 0; j < 16; j += 4) {
      v4f q = *(const v4f*)&R0[l * 16 + j];
      cost += q[0] + q[1] + q[2] + q[3];
    }
    float hk = sqrtf(fmaxf(cost, 0.f));
    float g = cf[(sbase + l) * 4 + 1];
    float mean = stats[0], off = stats[1];
    bool hv = stats[2] > 0.5f;
    bool valid = (g >= 0.2f) && (g <= 1.5f);
    float w;
    if (hv && valid) w = (g >= mean + off) ? 1.0f : ((g <= mean - off) ? 0.3f : 0.7f);
    else             w = 0.3f;
    whk = w * hk; ww = w;
  }
#pragma unroll
  for (int m = 1; m < 16; m <<= 1) {
    whk += __shfl_xor(whk, m, 32);
    ww  += __shfl_xor(ww,  m, 32);
  }
  if (l == 0) { wpart[2 * gw] = whk; wpart[2 * gw + 1] = ww; }
}

// ---------------- kernel 5: deterministic final reduction ----------------
__global__ void k_final(const float* __restrict__ wpart, int nwaves, float* __restrict__ out) {
  __shared__ float s0[1024], s1[1024];
  int tid = threadIdx.x;
  float a = 0.f, b = 0.f;
  for (int i = tid; i < nwaves; i += 1024) { a += wpart[2 * i]; b += wpart[2 * i + 1]; }
  s0[tid] = a; s1[tid] = b;
  __syncthreads();
  for (int o = 512; o > 0; o >>= 1) {
    if (tid < o) { s0[tid] += s0[tid + o]; s1[tid] += s1[tid + o]; }
    __syncthreads();
  }
  if (tid == 0) out[0] = s0[0] / (s1[0] + EPSF);
}

extern "C" void kernel_launch(void* const* d_in, const int* in_sizes, int n_in,
                              void* d_out, int out_size, void* d_ws, size_t ws_size,
                              hipStream_t stream) {
  const float* pred   = (const float*)d_in[0];
  const float* target = (const float*)d_in[1];
  const float* cf     = (const float*)d_in[2];
  int B = in_sizes[0];

  float* ws    = (float*)d_ws;
  float* Kg    = ws;                       // 48*48
  float* KCg   = Kg + NPAD * NPAD;         // 48*48
  float* bpart = KCg + NPAD * NPAD;        // 3 * nb1
  int nb1      = (B + 255) / 256;
  float* stats = bpart + 3 * nb1;          // 3 (+pad)
  float* wpart = stats + 8;                // 2 * nwaves
  int nwaves   = B / 16;                   // B=131072 -> 8192 waves
  int nblocks  = nwaves / 8;               // 8 waves per 256-thread block

  k_setup <<<(NPAD * NPAD + 255) / 256, 256, 0, stream>>>(Kg, KCg);
  k_stats1<<<nb1, 256, 0, stream>>>(cf, B, bpart);
  k_stats2<<<1, 256, 0, stream>>>(bpart, nb1, stats);
  k_main  <<<nblocks, 256, 0, stream>>>(pred, target, cf, Kg, KCg, stats, wpart);
  k_final <<<1, 1024, 0, stream>>>(wpart, nwaves, (float*)d_out);
}